// PmeModule_6811818132235
// MI455X (gfx1250) — compile-verified
//
#include <hip/hip_runtime.h>
#include <hip/hip_bf16.h>
#include <math.h>

typedef __attribute__((ext_vector_type(2))) float v2f;
typedef __attribute__((ext_vector_type(8))) float v8f;

#define N_ATOMS   4096
#define NTILE     256            // 4096 / 16
#define KXg       14
#define KYg       15
#define KZg       16
#define NG        (KXg*KYg*KZg)  // 3360
#define ALPHA_C   4.985823141035867f
#define COULOMB_C 138.935f
#define CUTOFF_C  0.5f
#define PI_F      3.14159265358979323846f

// ---------------------------------------------------------------------------
// helpers
// ---------------------------------------------------------------------------
__device__ __forceinline__ float fast_erfc(float x) {
    // Abramowitz & Stegun 7.1.26, |err| < 1.5e-7, x >= 0
    float t = __builtin_amdgcn_rcpf(1.0f + 0.3275911f * x);
    float p = t * (0.254829592f +
              t * (-0.284496736f +
              t * (1.421413741f +
              t * (-1.453152027f +
              t * 1.061405429f))));
    return p * __expf(-x * x);
}

// cardinal B-splines M_k (Essmann et al. 1995)
__device__ __forceinline__ float M1f(float x) { return (x >= 0.0f && x < 1.0f) ? 1.0f : 0.0f; }
__device__ __forceinline__ float M2f(float x) { return x * M1f(x) + (2.0f - x) * M1f(x - 1.0f); }
__device__ __forceinline__ float M3f(float x) { return (x * M2f(x) + (3.0f - x) * M2f(x - 1.0f)) * 0.5f; }
__device__ __forceinline__ float M4f(float x) { return (x * M3f(x) + (4.0f - x) * M3f(x - 1.0f)) * (1.0f/3.0f); }
__device__ __forceinline__ float M5f(float x) { return (x * M4f(x) + (5.0f - x) * M4f(x - 1.0f)) * 0.25f; }

__device__ __forceinline__ float bmod_axis(int m, int K) {
    // |b(m)|^-2 for order-5 splines; M5 at integers 1..4 = {1,11,11,1}/24
    float Mv[4] = {1.0f/24.0f, 11.0f/24.0f, 11.0f/24.0f, 1.0f/24.0f};
    float re = 0.0f, im = 0.0f;
    #pragma unroll
    for (int k = 0; k < 4; ++k) {
        float ang = 2.0f * PI_F * (float)m * (float)k / (float)K;
        float s, c;
        __sincosf(ang, &s, &c);
        re += Mv[k] * c;
        im += Mv[k] * s;
    }
    float d2 = re * re + im * im;
    return (d2 < 1e-7f) ? 0.0f : __builtin_amdgcn_rcpf(d2);
}

// ---------------------------------------------------------------------------
// workspace init: zero accumulator + charge grid
// ---------------------------------------------------------------------------
__global__ void init_kernel(float* ws) {
    int i = blockIdx.x * blockDim.x + threadIdx.x;
    if (i < 16 + NG) ws[i] = 0.0f;
}

// ---------------------------------------------------------------------------
// spread charges onto mesh with order-5 B-splines (periodic, atomics)
// ---------------------------------------------------------------------------
__global__ void spread_kernel(const float* __restrict__ pos,
                              const float* __restrict__ q,
                              const float* __restrict__ box,
                              float* __restrict__ grid) {
    int i = blockIdx.x * blockDim.x + threadIdx.x;
    if (i >= N_ATOMS) return;
    float L[3] = {box[0], box[4], box[8]};
    const int K[3] = {KXg, KYg, KZg};
    float w[3][5];
    int   idx[3][5];
    #pragma unroll
    for (int a = 0; a < 3; ++a) {
        float f = pos[i * 3 + a] / L[a];
        f -= floorf(f);
        float u = f * (float)K[a];
        float b = floorf(u);
        float ww = u - b;
        int bi = (int)b;
        #pragma unroll
        for (int k = 0; k < 5; ++k) {
            w[a][k] = M5f(ww + (float)k);
            int id = bi - k;
            id %= K[a];
            if (id < 0) id += K[a];
            idx[a][k] = id;
        }
    }
    float qi = q[i];
    #pragma unroll 1
    for (int ix = 0; ix < 5; ++ix) {
        #pragma unroll 1
        for (int iy = 0; iy < 5; ++iy) {
            float wxy = qi * w[0][ix] * w[1][iy];
            int base = idx[0][ix] * (KYg * KZg) + idx[1][iy] * KZg;
            #pragma unroll
            for (int iz = 0; iz < 5; ++iz) {
                atomicAdd(&grid[base + idx[2][iz]], wxy * w[2][iz]);
            }
        }
    }
}

// ---------------------------------------------------------------------------
// separable DFT stage along z:  Fz[x,y,mz] = sum_z Q[x,y,z] e^{-2pi i mz z/KZ}
// ---------------------------------------------------------------------------
__global__ void stagez_kernel(const float* __restrict__ grid,
                              float* __restrict__ fre, float* __restrict__ fim) {
    int id = blockIdx.x * blockDim.x + threadIdx.x;
    if (id >= NG) return;
    int x  = id / (KYg * KZg);
    int y  = (id / KZg) % KYg;
    int mz = id % KZg;
    float re = 0.0f, im = 0.0f;
    #pragma unroll 1
    for (int z = 0; z < KZg; ++z) {
        float qv = grid[x * (KYg * KZg) + y * KZg + z];
        float ang = -2.0f * PI_F * (float)(mz * z) / (float)KZg;
        float s, c;
        __sincosf(ang, &s, &c);
        re += qv * c;
        im += qv * s;
    }
    fre[id] = re;
    fim[id] = im;
}

// stage along y: Fyz[x,my,mz] = sum_y Fz[x,y,mz] e^{-2pi i my y/KY}
__global__ void stagey_kernel(const float* __restrict__ zre, const float* __restrict__ zim,
                              float* __restrict__ fre, float* __restrict__ fim) {
    int id = blockIdx.x * blockDim.x + threadIdx.x;
    if (id >= NG) return;
    int x  = id / (KYg * KZg);
    int my = (id / KZg) % KYg;
    int mz = id % KZg;
    float re = 0.0f, im = 0.0f;
    #pragma unroll 1
    for (int y = 0; y < KYg; ++y) {
        int src = x * (KYg * KZg) + y * KZg + mz;
        float ar = zre[src], ai = zim[src];
        float ang = -2.0f * PI_F * (float)(my * y) / (float)KYg;
        float s, c;
        __sincosf(ang, &s, &c);
        re += ar * c - ai * s;
        im += ar * s + ai * c;
    }
    fre[id] = re;
    fim[id] = im;
}

// stage along x + mode energy:  acc -= C/(2 pi V) * green * B * |F|^2
__global__ void modes_kernel(const float* __restrict__ yre, const float* __restrict__ yim,
                             const float* __restrict__ box, float* __restrict__ acc) {
    int id = blockIdx.x * blockDim.x + threadIdx.x;
    if (id >= NG) return;
    int mx = id / (KYg * KZg);
    int my = (id / KZg) % KYg;
    int mz = id % KZg;
    float re = 0.0f, im = 0.0f;
    #pragma unroll 1
    for (int x = 0; x < KXg; ++x) {
        int src = x * (KYg * KZg) + my * KZg + mz;
        float ar = yre[src], ai = yim[src];
        float ang = -2.0f * PI_F * (float)(mx * x) / (float)KXg;
        float s, c;
        __sincosf(ang, &s, &c);
        re += ar * c - ai * s;
        im += ar * s + ai * c;
    }
    float B = bmod_axis(mx, KXg) * bmod_axis(my, KYg) * bmod_axis(mz, KZg);
    float fx = (mx >= (KXg + 1) / 2) ? (float)(mx - KXg) : (float)mx;
    float fy = (my >= (KYg + 1) / 2) ? (float)(my - KYg) : (float)my;
    float fz = (mz >= (KZg + 1) / 2) ? (float)(mz - KZg) : (float)mz;
    float Lx = box[0], Ly = box[4], Lz = box[8];
    float vx = fx / Lx, vy = fy / Ly, vz = fz / Lz;
    float m2 = vx * vx + vy * vy + vz * vz;
    float e = 0.0f;
    if (m2 > 0.0f) {
        float green = __expf(-(PI_F * PI_F) * m2 / (ALPHA_C * ALPHA_C)) / m2;
        float V = Lx * Ly * Lz;
        e = (COULOMB_C / (2.0f * PI_F * V)) * green * B * (re * re + im * im);
    }
    atomicAdd(acc, -e);   // output = edir - erecip
}

// self energy: output gets -eself = +C*alpha/sqrt(pi) * sum q^2
__global__ void self_kernel(const float* __restrict__ q, float* __restrict__ acc) {
    int t = threadIdx.x;
    float s = 0.0f;
    for (int i = t; i < N_ATOMS; i += 256) {
        float qi = q[i];
        s += qi * qi;
    }
    for (int off = 16; off > 0; off >>= 1) s += __shfl_down(s, off, 32);
    if ((t & 31) == 0) atomicAdd(acc, (COULOMB_C * ALPHA_C / sqrtf(PI_F)) * s);
}

// ---------------------------------------------------------------------------
// direct-space sum.
//  * One block (256 threads, 8 waves) owns one 16-row band of the pair matrix.
//  * Block stages ALL positions (48KB) + charges (16KB) into LDS with the
//    CDNA5 async copy path (global_load_async_to_lds_b128 + s_wait_asynccnt),
//    then every A/B fragment read is an LDS read.
//  * Each wave sweeps 32 column tiles: per tile one V_WMMA_F32_16X16X4_F32
//    produces the 16x16 Gram matrix; epilogue applies the minimum-image wrap
//    correction and a branch-guarded erfc using v_rsq/v_rcp/v_exp.
// ---------------------------------------------------------------------------
__global__ void __launch_bounds__(256)
direct_kernel(const float* __restrict__ pos,
              const float* __restrict__ q,
              const float* __restrict__ box,
              float* __restrict__ acc) {
    __shared__ float spos[N_ATOMS * 3];   // 48 KB
    __shared__ float sq[N_ATOMS];         // 16 KB   (total 64 KB of 320 KB/WGP)

    int t = threadIdx.x;

    // ---- async stage of positions: 3072 x 16B chunks, 12 per thread ----
    {
        unsigned long long gp = (unsigned long long)(const void*)pos;
        #pragma unroll 1
        for (int c = t; c < (N_ATOMS * 3) / 4; c += 256) {
            unsigned int  loff = (unsigned int)(unsigned long long)(const void*)&spos[c * 4];
            unsigned long long ga = gp + (unsigned long long)c * 16ull;
            asm volatile("global_load_async_to_lds_b128 %0, %1, off"
                         :: "v"(loff), "v"(ga) : "memory");
        }
        unsigned long long gq = (unsigned long long)(const void*)q;
        #pragma unroll 1
        for (int c = t; c < N_ATOMS / 4; c += 256) {
            unsigned int  loff = (unsigned int)(unsigned long long)(const void*)&sq[c * 4];
            unsigned long long ga = gq + (unsigned long long)c * 16ull;
            asm volatile("global_load_async_to_lds_b128 %0, %1, off"
                         :: "v"(loff), "v"(ga) : "memory");
        }
        asm volatile("s_wait_asynccnt 0x0" ::: "memory");
    }
    __syncthreads();

    int wave = t >> 5;
    int lane = t & 31;
    int I    = blockIdx.x;        // row tile (0..255)
    int half = lane >> 4;         // 0 or 1
    int l15  = lane & 15;

    float Lx = box[0], Ly = box[4], Lz = box[8];
    float invLx = __builtin_amdgcn_rcpf(Lx);
    float invLy = __builtin_amdgcn_rcpf(Ly);
    float invLz = __builtin_amdgcn_rcpf(Lz);

    // A-matrix 16x4 f32 layout: VGPR0 = K={0,2}, VGPR1 = K={1,3}; K=3 pad = 0
    int arow = I * 16 + l15;
    v2f a;
    a.x = spos[arow * 3 + (half ? 2 : 0)];
    a.y = half ? 0.0f : spos[arow * 3 + 1];

    // lane-resident row data (lane l holds row l of the band)
    float rx = spos[arow * 3 + 0];
    float ry = spos[arow * 3 + 1];
    float rz = spos[arow * 3 + 2];
    float rq = sq[arow];
    float rq2 = rx * rx + ry * ry + rz * rz;

    // hoist the row broadcasts: they depend only on the row band, not on J
    float rowX[8], rowY[8], rowZ[8], rowQ[8], rowQ2[8];
    int   rowIdx[8];
    #pragma unroll
    for (int v = 0; v < 8; ++v) {
        int m = v + (half << 3);                 // C/D layout: row = v (+8 hi lanes)
        rowX[v]  = __shfl(rx,  m, 32);
        rowY[v]  = __shfl(ry,  m, 32);
        rowZ[v]  = __shfl(rz,  m, 32);
        rowQ[v]  = __shfl(rq,  m, 32);
        rowQ2[v] = __shfl(rq2, m, 32);
        rowIdx[v] = I * 16 + m;
    }

    float e = 0.0f;
    const float c2 = CUTOFF_C * CUTOFF_C;

    #pragma unroll 2
    for (int J = wave; J < NTILE; J += 8) {
        int bcol = J * 16 + l15;
        // B-matrix 4x16: lane = column, same K striping as A
        v2f b;
        b.x = spos[bcol * 3 + (half ? 2 : 0)];
        b.y = half ? 0.0f : spos[bcol * 3 + 1];

        v8f cz = {};
        // G[m][n] = p_i . p_j  (16x16 tile Gram matrix)
        v8f g = __builtin_amdgcn_wmma_f32_16x16x4_f32(
            /*neg_a=*/false, a, /*neg_b=*/false, b,
            /*c_mod=*/(short)0, cz, /*reuse_a=*/false, /*reuse_b=*/false);

        float cxv = spos[bcol * 3 + 0];
        float cyv = spos[bcol * 3 + 1];
        float czv = spos[bcol * 3 + 2];
        float cq  = sq[bcol];
        float cq2 = cxv * cxv + cyv * cyv + czv * czv;

        #pragma unroll
        for (int v = 0; v < 8; ++v) {
            float r20 = rowQ2[v] + cq2 - 2.0f * g[v];   // unwrapped r^2 (WMMA)
            float dx = rowX[v] - cxv;
            float dy = rowY[v] - cyv;
            float dz = rowZ[v] - czv;
            float nx = rintf(dx * invLx);
            float ny = rintf(dy * invLy);
            float nz = rintf(dz * invLz);
            float r2 = r20 + nx * Lx * (nx * Lx - 2.0f * dx)
                           + ny * Ly * (ny * Ly - 2.0f * dy)
                           + nz * Lz * (nz * Lz - 2.0f * dz);
            if (r2 < c2 && rowIdx[v] != bcol) {
                float rinv = __builtin_amdgcn_rsqf(r2);   // v_rsq_f32
                float r = r2 * rinv;
                e += rowQ[v] * cq * fast_erfc(ALPHA_C * r) * rinv;
            }
        }
    }

    for (int off = 16; off > 0; off >>= 1) e += __shfl_down(e, off, 32);
    if (lane == 0) atomicAdd(acc, 0.5f * COULOMB_C * e);
}

// ---------------------------------------------------------------------------
__global__ void finalize_kernel(const float* __restrict__ acc, float* __restrict__ out) {
    out[0] = acc[0];
}

extern "C" void kernel_launch(void* const* d_in, const int* in_sizes, int n_in,
                              void* d_out, int out_size, void* d_ws, size_t ws_size,
                              hipStream_t stream) {
    const float* pos = (const float*)d_in[0];   // [4096,3] f32
    const float* q   = (const float*)d_in[1];   // [4096]   f32
    const float* box = (const float*)d_in[2];   // [3,3]    f32 (diagonal)

    float* ws    = (float*)d_ws;
    float* acc   = ws;                // [0]
    float* grid  = ws + 16;           // [NG]
    float* fz_re = grid + NG;         // [NG]
    float* fz_im = fz_re + NG;        // [NG]
    float* fy_re = fz_im + NG;        // [NG]
    float* fy_im = fy_re + NG;        // [NG]
    float* out   = (float*)d_out;

    init_kernel<<<dim3((16 + NG + 511) / 512), dim3(512), 0, stream>>>(ws);
    spread_kernel<<<dim3(N_ATOMS / 256), dim3(256), 0, stream>>>(pos, q, box, grid);
    stagez_kernel<<<dim3((NG + 255) / 256), dim3(256), 0, stream>>>(grid, fz_re, fz_im);
    stagey_kernel<<<dim3((NG + 255) / 256), dim3(256), 0, stream>>>(fz_re, fz_im, fy_re, fy_im);
    modes_kernel<<<dim3((NG + 255) / 256), dim3(256), 0, stream>>>(fy_re, fy_im, box, acc);
    self_kernel<<<dim3(1), dim3(256), 0, stream>>>(q, acc);
    direct_kernel<<<dim3(NTILE), dim3(256), 0, stream>>>(pos, q, box, acc);
    finalize_kernel<<<dim3(1), dim3(1), 0, stream>>>(acc, out);
}